// DeformableConvBlock_show_79233556676780
// MI455X (gfx1250) — compile-verified
//
#include <hip/hip_runtime.h>

// DeformableConvBlock for MI455X (gfx1250), fp32 end-to-end via
// V_WMMA_F32_16X16X4_F32.
//
// - K is tap-major (k = tap*CIN + c); 3x3 tap loop fully unrolled, border
//   masks / clamped offsets / deform bilinear metadata hoisted per tap.
// - Weights pre-swizzled into per-lane WMMA fragment order
//   [kstep][couttile][lane][2] so every A load is one coalesced 256B b64.
// - Each wave computes M=32 pixels (2 B fragments) x COUT_PAD, reusing every
//   A fragment twice: conv1 inner step = 8 A loads + 4 B loads per 16 WMMA.
//
// Scratch (d_ws, floats):
//   est   [8,128,128,128]  16777216
//   offt  [8, 18,128,128]   2359296
//   wsEst (1152/4)*8*64      147456
//   wsOff (1152/4)*2*64       36864   (couts 18..31 zero)
//   wsDef ( 576/4)*4*64       36864
//   total ~77.5 MB

typedef float v2f __attribute__((ext_vector_type(2)));
typedef float v8f __attribute__((ext_vector_type(8)));

#define BN 8
#define CBASE 64
#define HH 128
#define WW 128
#define HWSZ (HH * WW)

__device__ __forceinline__ int iclamp(int v, int lo, int hi) {
    return v < lo ? lo : (v > hi ? hi : v);
}

// ---------------------------------------------------------------------------
// Weight swizzle: OIHW [COUT][CIN][3][3] -> fragment-major
//   wsw[((s*NT + t)*32 + lane)*2 + v] = w[cout = t*16+lane%16][k = s*4 + (lane/16)*2 + v]
// with k = tap*CIN + c (tap-major K), zero-padded couts.
// ---------------------------------------------------------------------------
template <int CIN, int COUT, int COUT_PAD>
__global__ __launch_bounds__(256) void swizzle_w_kernel(
    const float* __restrict__ w, float* __restrict__ wsw)
{
    constexpr int K  = CIN * 9;
    constexpr int NT = COUT_PAD / 16;
    const int i = blockIdx.x * 256 + threadIdx.x;
    if (i >= (K / 4) * NT * 64) return;
    const int v    = i & 1;
    const int l    = (i >> 1) & 31;
    const int t    = (i >> 6) % NT;
    const int s    = i / (64 * NT);
    const int cout = t * 16 + (l & 15);
    const int k    = s * 4 + ((l >> 4) << 1) + v;
    const int tap  = k / CIN;
    const int c    = k - tap * CIN;
    wsw[i] = (cout < COUT) ? w[((size_t)cout * CIN + c) * 9 + tap] : 0.f;
}

// ---------------------------------------------------------------------------
// 3x3 conv, stride 1, pad 1, implicit GEMM on f32 WMMA, tap-major K,
// fragment-swizzled weights, M=32 pixels per wave (2 B fragments).
// ---------------------------------------------------------------------------
template <int CIN, int COUT, int COUT_PAD, bool CONCAT, bool LEAKY>
__global__ __launch_bounds__(128) void conv3x3_wmma_kernel(
    const float* __restrict__ xa,   // first half channels (or whole input)
    const float* __restrict__ xb,   // second half channels if CONCAT
    const float* __restrict__ wsw,  // fragment-swizzled weights
    float* __restrict__ out)        // [B, COUT, H, W]
{
    constexpr int NT  = COUT_PAD / 16;
    constexpr int NCH = CONCAT ? (CIN >> 1) : CIN;   // channels per source
    constexpr int NH  = CONCAT ? 2 : 1;

    const int lane  = threadIdx.x & 31;
    const int wave  = threadIdx.x >> 5;
    const int gtile = blockIdx.x * 4 + wave;   // 32-pixel group index

    const int pbase = gtile * 32;
    const int b     = pbase / HWSZ;
    const int yx0   = pbase - b * HWSZ;
    const int y     = yx0 / WW;
    const int xbase = yx0 - y * WW;            // 32 consecutive x, same row
    const int x0    = xbase + (lane & 15);     // tile 0 pixel x
    const int x1    = x0 + 16;                 // tile 1 pixel x

    const int khalf = (lane >> 4) << 1;        // lanes 16..31 hold K+2,K+3
    const float* wp = wsw + (size_t)lane * 2;  // fragment-major A base

    v8f acc0[NT], acc1[NT];
#pragma unroll
    for (int t = 0; t < NT; ++t) {
        acc0[t] = v8f{0.f, 0.f, 0.f, 0.f, 0.f, 0.f, 0.f, 0.f};
        acc1[t] = v8f{0.f, 0.f, 0.f, 0.f, 0.f, 0.f, 0.f, 0.f};
    }

#pragma unroll
    for (int tap = 0; tap < 9; ++tap) {
        const int dy = tap / 3 - 1;
        const int dx = tap % 3 - 1;
        const int yy = y + dy;
        const bool oky = (yy >= 0) && (yy < HH);
        const int yyc  = iclamp(yy, 0, HH - 1);
        // per-pixel-tile border masks + clamped spatial offsets
        const int xx0 = x0 + dx, xx1 = x1 + dx;
        const float msk0 = (oky && xx0 >= 0 && xx0 < WW) ? 1.f : 0.f;
        const float msk1 = (oky && xx1 >= 0 && xx1 < WW) ? 1.f : 0.f;
        const int spat0 = yyc * WW + iclamp(xx0, 0, WW - 1);
        const int spat1 = yyc * WW + iclamp(xx1, 0, WW - 1);

#pragma unroll
        for (int h = 0; h < NH; ++h) {
            const float* src = (CONCAT && h) ? xb : xa;
            const float* sp0 = src + ((size_t)b * NCH + khalf) * HWSZ + spat0;
            const float* sp1 = src + ((size_t)b * NCH + khalf) * HWSZ + spat1;
            const int s0 = tap * (CIN / 4) + h * (NCH / 4);  // K-step base

            for (int cb = 0; cb < NCH; cb += 4) {
                v2f b0, b1;
                b0[0] = msk0 * sp0[(size_t)cb * HWSZ];
                b0[1] = msk0 * sp0[(size_t)(cb + 1) * HWSZ];
                b1[0] = msk1 * sp1[(size_t)cb * HWSZ];
                b1[1] = msk1 * sp1[(size_t)(cb + 1) * HWSZ];
                const float* wk = wp + (size_t)(s0 + (cb >> 2)) * NT * 64;
#pragma unroll
                for (int t = 0; t < NT; ++t) {
                    const v2f afrag = *(const v2f*)(wk + t * 64);
                    acc0[t] = __builtin_amdgcn_wmma_f32_16x16x4_f32(
                        false, afrag, false, b0, (short)0, acc0[t], false, false);
                    acc1[t] = __builtin_amdgcn_wmma_f32_16x16x4_f32(
                        false, afrag, false, b1, (short)0, acc1[t], false, false);
                }
            }
        }
    }

    // store D tiles: row M = cout = t*16 + 8*(lane>=16) + r, col N = pixel
    const int mhalf = (lane >> 4) << 3;
    const int yxs0  = y * WW + x0;
    const int yxs1  = y * WW + x1;
#pragma unroll
    for (int t = 0; t < NT; ++t) {
#pragma unroll
        for (int r = 0; r < 8; ++r) {
            const int cout = t * 16 + mhalf + r;
            float v0 = acc0[t][r], v1 = acc1[t][r];
            if (LEAKY) {
                v0 = v0 > 0.f ? v0 : 0.1f * v0;
                v1 = v1 > 0.f ? v1 : 0.1f * v1;
            }
            if (COUT == COUT_PAD || cout < COUT) {
                out[((size_t)b * COUT + cout) * HWSZ + yxs0] = v0;
                out[((size_t)b * COUT + cout) * HWSZ + yxs1] = v1;
            }
        }
    }
}

// ---------------------------------------------------------------------------
// Deformable conv 3x3 (1 offset group, no bias), tap-major K = tap*64 + c,
// fragment-swizzled weights, M=32 pixels per wave.
// Phase 1: bilinear metadata (32 px x 9 taps) -> LDS (b128 DS ops)
// Phase 2: tap loop (metadata hoisted per tap) x channel GEMM on WMMA.
// ---------------------------------------------------------------------------
__global__ __launch_bounds__(128) void deform_conv_wmma_kernel(
    const float* __restrict__ xin,  // hr [B,64,H,W]
    const float* __restrict__ off,  // [B,18,H,W] interleaved (dy,dx) per tap
    const float* __restrict__ wsw,  // fragment-swizzled w_def
    float* __restrict__ out)        // [B,64,H,W]
{
    constexpr int NT = CBASE / 16;  // 4

    __shared__ int4   sidx[4 * 288];
    __shared__ float4 swts[4 * 288];

    const int lane  = threadIdx.x & 31;
    const int wave  = threadIdx.x >> 5;
    const int gtile = blockIdx.x * 4 + wave;
    const int pbase = gtile * 32;
    const int b     = pbase / HWSZ;
    const int yx0   = pbase - b * HWSZ;
    const int y     = yx0 / WW;
    const int xbase = yx0 - y * WW;   // 32 consecutive x, same row

    // ---- phase 1: bilinear corner indices + masked weights into LDS ----
    for (int i = lane; i < 288; i += 32) {
        const int m   = i / 9;         // pixel 0..31
        const int tap = i - m * 9;
        const int x   = xbase + m;
        const int t3  = tap / 3;
        const int dy  = t3 - 1;
        const int dx  = tap - t3 * 3 - 1;
        const int yxp = y * WW + x;
        const float ody = off[(((size_t)b * 9 + tap) * 2 + 0) * HWSZ + yxp];
        const float odx = off[(((size_t)b * 9 + tap) * 2 + 1) * HWSZ + yxp];
        const float py  = ody + (float)(y + dy);
        const float px  = odx + (float)(x + dx);
        const float fy0 = floorf(py);
        const float fx0 = floorf(px);
        const float wy1 = py - fy0, wx1 = px - fx0;
        const float wy0 = 1.f - wy1, wx0 = 1.f - wx1;
        const int yi = (int)fy0, xi = (int)fx0;

        int4 ci; float4 cw;
        {   const int yy = yi, xx = xi;
            const bool ok = (yy >= 0) && (yy < HH) && (xx >= 0) && (xx < WW);
            ci.x = iclamp(yy, 0, HH - 1) * WW + iclamp(xx, 0, WW - 1);
            cw.x = ok ? wy0 * wx0 : 0.f; }
        {   const int yy = yi, xx = xi + 1;
            const bool ok = (yy >= 0) && (yy < HH) && (xx >= 0) && (xx < WW);
            ci.y = iclamp(yy, 0, HH - 1) * WW + iclamp(xx, 0, WW - 1);
            cw.y = ok ? wy0 * wx1 : 0.f; }
        {   const int yy = yi + 1, xx = xi;
            const bool ok = (yy >= 0) && (yy < HH) && (xx >= 0) && (xx < WW);
            ci.z = iclamp(yy, 0, HH - 1) * WW + iclamp(xx, 0, WW - 1);
            cw.z = ok ? wy1 * wx0 : 0.f; }
        {   const int yy = yi + 1, xx = xi + 1;
            const bool ok = (yy >= 0) && (yy < HH) && (xx >= 0) && (xx < WW);
            ci.w = iclamp(yy, 0, HH - 1) * WW + iclamp(xx, 0, WW - 1);
            cw.w = ok ? wy1 * wx1 : 0.f; }

        sidx[wave * 288 + i] = ci;
        swts[wave * 288 + i] = cw;
    }
    __syncthreads();

    // ---- phase 2: tap-major WMMA GEMM with gathered B fragments ----
    const int khalf = (lane >> 4) << 1;
    const int m0    = lane & 15;
    const float* wp = wsw + (size_t)lane * 2;

    v8f acc0[NT], acc1[NT];
#pragma unroll
    for (int t = 0; t < NT; ++t) {
        acc0[t] = v8f{0.f, 0.f, 0.f, 0.f, 0.f, 0.f, 0.f, 0.f};
        acc1[t] = v8f{0.f, 0.f, 0.f, 0.f, 0.f, 0.f, 0.f, 0.f};
    }

    const float* xc = xin + ((size_t)b * CBASE + khalf) * HWSZ;

#pragma unroll
    for (int tap = 0; tap < 9; ++tap) {
        const int4   ci0 = sidx[wave * 288 + m0 * 9 + tap];          // tile 0
        const float4 cw0 = swts[wave * 288 + m0 * 9 + tap];
        const int4   ci1 = sidx[wave * 288 + (m0 + 16) * 9 + tap];   // tile 1
        const float4 cw1 = swts[wave * 288 + (m0 + 16) * 9 + tap];
        const int s0 = tap * (CBASE / 4);

        for (int cb = 0; cb < CBASE; cb += 4) {
            const float* c0 = xc + (size_t)cb * HWSZ;
            const float* c1 = c0 + HWSZ;
            v2f b0, b1;
            b0[0] = cw0.x * c0[ci0.x] + cw0.y * c0[ci0.y]
                  + cw0.z * c0[ci0.z] + cw0.w * c0[ci0.w];
            b0[1] = cw0.x * c1[ci0.x] + cw0.y * c1[ci0.y]
                  + cw0.z * c1[ci0.z] + cw0.w * c1[ci0.w];
            b1[0] = cw1.x * c0[ci1.x] + cw1.y * c0[ci1.y]
                  + cw1.z * c0[ci1.z] + cw1.w * c0[ci1.w];
            b1[1] = cw1.x * c1[ci1.x] + cw1.y * c1[ci1.y]
                  + cw1.z * c1[ci1.z] + cw1.w * c1[ci1.w];
            const float* wk = wp + (size_t)(s0 + (cb >> 2)) * NT * 64;
#pragma unroll
            for (int t = 0; t < NT; ++t) {
                const v2f afrag = *(const v2f*)(wk + t * 64);
                acc0[t] = __builtin_amdgcn_wmma_f32_16x16x4_f32(
                    false, afrag, false, b0, (short)0, acc0[t], false, false);
                acc1[t] = __builtin_amdgcn_wmma_f32_16x16x4_f32(
                    false, afrag, false, b1, (short)0, acc1[t], false, false);
            }
        }
    }

    const int mhalf = (lane >> 4) << 3;
    const int yxs0  = y * WW + xbase + (lane & 15);
#pragma unroll
    for (int t = 0; t < NT; ++t) {
#pragma unroll
        for (int r = 0; r < 8; ++r) {
            const int cout = t * 16 + mhalf + r;
            out[((size_t)b * CBASE + cout) * HWSZ + yxs0]      = acc0[t][r];
            out[((size_t)b * CBASE + cout) * HWSZ + yxs0 + 16] = acc1[t][r];
        }
    }
}

extern "C" void kernel_launch(void* const* d_in, const int* in_sizes, int n_in,
                              void* d_out, int out_size, void* d_ws, size_t ws_size,
                              hipStream_t stream) {
    (void)in_sizes; (void)n_in; (void)out_size; (void)ws_size;

    const float* lr    = (const float*)d_in[0];  // [8,64,128,128]
    const float* hr    = (const float*)d_in[1];  // [8,64,128,128]
    const float* w_est = (const float*)d_in[2];  // [128,128,3,3]
    const float* w_off = (const float*)d_in[3];  // [18,128,3,3]
    const float* w_def = (const float*)d_in[4];  // [64,64,3,3]

    float* est   = (float*)d_ws;                          // 16777216
    float* offt  = est   + (size_t)BN * 128 * HWSZ;       //  2359296
    float* wsEst = offt  + (size_t)BN * 18 * HWSZ;        //   147456
    float* wsOff = wsEst + 147456;                        //    36864
    float* wsDef = wsOff + 36864;                         //    36864
    float* out   = (float*)d_out;

    // one-time weight swizzles (tiny)
    swizzle_w_kernel<128, 128, 128>
        <<<(147456 + 255) / 256, 256, 0, stream>>>(w_est, wsEst);
    swizzle_w_kernel<128, 18, 32>
        <<<(36864 + 255) / 256, 256, 0, stream>>>(w_off, wsOff);
    swizzle_w_kernel<64, 64, 64>
        <<<(36864 + 255) / 256, 256, 0, stream>>>(w_def, wsDef);

    const int groups = BN * HWSZ / 32;  // 4096 32-pixel groups
    const int blocks = groups / 4;      // 4 waves (groups) per block

    // Stage 1: concat + conv3x3(w_est) + leaky_relu(0.1)
    conv3x3_wmma_kernel<128, 128, 128, true, true>
        <<<blocks, 128, 0, stream>>>(lr, hr, wsEst, est);
    // Stage 2: conv3x3(w_off) -> 18 offset channels (padded to 32)
    conv3x3_wmma_kernel<128, 18, 32, false, false>
        <<<blocks, 128, 0, stream>>>(est, est, wsOff, offt);
    // Stage 3: deformable conv
    deform_conv_wmma_kernel<<<blocks, 128, 0, stream>>>(hr, offt, wsDef, out);
}